// GCN_28157805592771
// MI455X (gfx1250) — compile-verified
//
#include <hip/hip_runtime.h>
#include <hip/hip_bf16.h>

#define INDIM  256
#define HIDIM  128
#define OUTDIM 64
#define MBLK   5          // M-tiles (of 16 rows) per wave; 50000/16 = 3125 = 5*625

typedef __attribute__((ext_vector_type(2))) float v2f;
typedef __attribute__((ext_vector_type(8))) float v8f;

// ---------------------------------------------------------------------------
// Degree / normalization kernels
// ---------------------------------------------------------------------------
__global__ __launch_bounds__(256) void k_deg_init(int* deg, int n) {
    int i = blockIdx.x * blockDim.x + threadIdx.x;
    if (i < n) deg[i] = 1;  // self-loop contributes 1 to every node's in-degree
}

__global__ __launch_bounds__(256) void k_deg_count(const int* __restrict__ col,
                                                   int* __restrict__ deg, int e) {
    int i = blockIdx.x * blockDim.x + threadIdx.x;
    if (i < e) atomicAdd(&deg[col[i]], 1);
}

__global__ __launch_bounds__(256) void k_dinv(const int* __restrict__ deg,
                                              float* __restrict__ dinv, int n) {
    int i = blockIdx.x * blockDim.x + threadIdx.x;
    if (i < n) dinv[i] = rsqrtf((float)deg[i]);   // deg >= 1 always (self-loops)
}

// ---------------------------------------------------------------------------
// fp32 WMMA GEMM: C[M,N] = A[M,K] * B[K,N].
// One wave computes a (TB*16)x16 C macro-tile = TB stacked 16x16 WMMA tiles
// that SHARE the B operand. GUARD-FREE hot path: per K-step,
//   TB x global_load_b64 (A) + 2 x b32 (B)  ->  TB x v_wmma_f32_16x16x4_f32.
// M-remainder tiles are handled by a separate TB=1 launch (mOffset), so the
// inner loop has zero branches / predication.
//
// V_WMMA_F32_16X16X4_F32 VGPR layouts (ISA 7.12.2):
//   A (16x4): lane l<16 holds A[l][0..1], lane 16+l holds A[l][2..3]  -> v2f
//   B (4x16): vgpr0 = B[2*half][n], vgpr1 = B[2*half+1][n], n = lane&15
//   C/D (16x16): vgpr r = C[r + 8*half][lane&15]                       -> v8f
// ---------------------------------------------------------------------------
template <int TB>
__global__ __launch_bounds__(32) void k_gemm_wmma_f32(const float* __restrict__ A,
                                                      const float* __restrict__ B,
                                                      float* __restrict__ C,
                                                      int mOffset, int N, int K) {
    const int lane  = threadIdx.x & 31;
    const int half  = lane >> 4;        // 0 or 1
    const int l     = lane & 15;
    const int mBase = mOffset + blockIdx.x * (TB * 16);
    const int nBase = blockIdx.y << 4;

    const float* ap = A + (size_t)(mBase + l) * K + (half << 1);   // K-pair per lane
    const float* bp = B + (size_t)(half << 1) * N + (nBase + l);
    const size_t aTileStride = (size_t)16 * K;                     // 16 rows down

    v8f acc[TB];
#pragma unroll
    for (int t = 0; t < TB; ++t) acc[t] = (v8f){};

#pragma unroll 2
    for (int k = 0; k < K; k += 4) {
        v2f b;
        b.x = bp[(size_t)k * N];
        b.y = bp[(size_t)k * N + N];
        v2f a[TB];
#pragma unroll
        for (int t = 0; t < TB; ++t)
            a[t] = *(const v2f*)(ap + aTileStride * t + k);        // global_load_b64
#pragma unroll
        for (int t = 0; t < TB; ++t)
            acc[t] = __builtin_amdgcn_wmma_f32_16x16x4_f32(
                /*neg_a=*/false, a[t], /*neg_b=*/false, b,
                /*c_mod=*/(short)0, acc[t], /*reuse_a=*/false, /*reuse_b=*/false);
    }

#pragma unroll
    for (int t = 0; t < TB; ++t) {
        float* cp = C + (size_t)(mBase + t * 16 + (half << 3)) * N + (nBase + l);
#pragma unroll
        for (int r = 0; r < 8; ++r) cp[(size_t)r * N] = acc[t][r];
    }
}

// ---------------------------------------------------------------------------
// Aggregation init (float4-vectorized): out[i,f] = h[i,f]*dinv[i]^2 + bias[f]
// Folds the self-loop edge and the bias into the accumulator init.
// ---------------------------------------------------------------------------
__global__ __launch_bounds__(256) void k_agg_init(const float4* __restrict__ h,
                                                  const float* __restrict__ dinv,
                                                  const float4* __restrict__ bias,
                                                  float4* __restrict__ out,
                                                  int total4, int F4) {
    int idx = blockIdx.x * blockDim.x + threadIdx.x;
    if (idx >= total4) return;
    int i = idx / F4, f = idx - i * F4;
    float d = dinv[i];
    float s = d * d;
    float4 v = h[idx], b = bias[f];
    out[idx] = make_float4(v.x * s + b.x, v.y * s + b.y, v.z * s + b.z, v.w * s + b.w);
}

// ---------------------------------------------------------------------------
// Edge scatter, F=128: one wave per edge, lane handles 4 contiguous features.
// Gather is a fully coalesced 512B read per wave; scatter via f32 atomics (L2).
// ---------------------------------------------------------------------------
__global__ __launch_bounds__(256) void k_scatter128(const int* __restrict__ row,
                                                    const int* __restrict__ col,
                                                    const float* __restrict__ dinv,
                                                    const float* __restrict__ h,
                                                    float* __restrict__ out, int E) {
    int e = blockIdx.x * (blockDim.x >> 5) + (threadIdx.x >> 5);
    if (e >= E) return;
    int lane = threadIdx.x & 31;
    int r = row[e], c = col[e];
    float norm = dinv[r] * dinv[c];
    float4 v = ((const float4*)(h + (size_t)r * 128))[lane];
    float* dst = out + (size_t)c * 128 + lane * 4;
    atomicAdd(dst + 0, v.x * norm);
    atomicAdd(dst + 1, v.y * norm);
    atomicAdd(dst + 2, v.z * norm);
    atomicAdd(dst + 3, v.w * norm);
}

// Edge scatter, F=64: one wave per edge, lane handles 2 contiguous features.
__global__ __launch_bounds__(256) void k_scatter64(const int* __restrict__ row,
                                                   const int* __restrict__ col,
                                                   const float* __restrict__ dinv,
                                                   const float* __restrict__ h,
                                                   float* __restrict__ out, int E) {
    int e = blockIdx.x * (blockDim.x >> 5) + (threadIdx.x >> 5);
    if (e >= E) return;
    int lane = threadIdx.x & 31;
    int r = row[e], c = col[e];
    float norm = dinv[r] * dinv[c];
    float2 v = ((const float2*)(h + (size_t)r * 64))[lane];
    float* dst = out + (size_t)c * 64 + lane * 2;
    atomicAdd(dst + 0, v.x * norm);
    atomicAdd(dst + 1, v.y * norm);
}

__global__ __launch_bounds__(256) void k_relu4(float4* __restrict__ a, int total4) {
    int i = blockIdx.x * blockDim.x + threadIdx.x;
    if (i < total4) {
        float4 v = a[i];
        a[i] = make_float4(fmaxf(v.x, 0.f), fmaxf(v.y, 0.f),
                           fmaxf(v.z, 0.f), fmaxf(v.w, 0.f));
    }
}

// ---------------------------------------------------------------------------
static void launch_gemm(const float* A, const float* B, float* C,
                        int M, int N, int K, hipStream_t stream) {
    const int tilesM = M / 16;            // M assumed multiple of 16 (50000 is)
    const int full   = tilesM / MBLK;     // macro-tiles with no guards
    const int rem    = tilesM - full * MBLK;
    if (full > 0)
        k_gemm_wmma_f32<MBLK><<<dim3(full, N / 16), 32, 0, stream>>>(
            A, B, C, 0, N, K);
    if (rem > 0)                          // zero for n=50000; kept for generality
        k_gemm_wmma_f32<1><<<dim3(rem, N / 16), 32, 0, stream>>>(
            A, B, C, full * MBLK * 16, N, K);
}

extern "C" void kernel_launch(void* const* d_in, const int* in_sizes, int n_in,
                              void* d_out, int out_size, void* d_ws, size_t ws_size,
                              hipStream_t stream) {
    const float* x    = (const float*)d_in[0];
    const int*   edge = (const int*)  d_in[1];
    const float* W1   = (const float*)d_in[2];
    const float* b1   = (const float*)d_in[3];
    const float* W2   = (const float*)d_in[4];
    const float* b2   = (const float*)d_in[5];
    float*       out  = (float*)d_out;

    const int n = in_sizes[0] / INDIM;   // 50000 (multiple of 16)
    const int E = in_sizes[1] / 2;       // 1600000
    const int* row = edge;               // edge_index[0] = sources
    const int* col = edge + E;           // edge_index[1] = destinations

    // Workspace carve-up (~64.5 MB total)
    char* w = (char*)d_ws;
    size_t off = 0;
    auto carve = [&](size_t bytes) {
        void* p = w + off;
        off += (bytes + 255) & ~(size_t)255;
        return p;
    };
    int*   deg  = (int*)  carve((size_t)n * 4);
    float* dinv = (float*)carve((size_t)n * 4);
    float* h1   = (float*)carve((size_t)n * HIDIM  * 4);
    float* agg1 = (float*)carve((size_t)n * HIDIM  * 4);
    float* h2   = (float*)carve((size_t)n * OUTDIM * 4);
    (void)ws_size;

    const int T = 256;
    const int wavesPerBlk = T / 32;

    // 1) degrees (self-loop baked in) and symmetric norm factors
    k_deg_init <<<(n + T - 1) / T, T, 0, stream>>>(deg, n);
    k_deg_count<<<(E + T - 1) / T, T, 0, stream>>>(col, deg, E);
    k_dinv     <<<(n + T - 1) / T, T, 0, stream>>>(deg, dinv, n);

    // 2) h1 = x @ W1   (fp32 WMMA, 80x16 guard-free macro-tile per wave)
    launch_gemm(x, W1, h1, n, HIDIM, INDIM, stream);

    // 3) agg1 = segment_sum(h1[row]*norm, col) + b1  (init = self-loop + bias)
    {
        int total4 = n * HIDIM / 4;
        k_agg_init  <<<(total4 + T - 1) / T, T, 0, stream>>>(
            (const float4*)h1, dinv, (const float4*)b1, (float4*)agg1, total4, HIDIM / 4);
        k_scatter128<<<(E + wavesPerBlk - 1) / wavesPerBlk, T, 0, stream>>>(
            row, col, dinv, h1, agg1, E);
        k_relu4     <<<(total4 + T - 1) / T, T, 0, stream>>>((float4*)agg1, total4);
    }

    // 4) h2 = relu(agg1) @ W2   (fp32 WMMA)
    launch_gemm(agg1, W2, h2, n, OUTDIM, HIDIM, stream);

    // 5) out = segment_sum(h2[row]*norm, col) + b2
    {
        int total4 = n * OUTDIM / 4;
        k_agg_init <<<(total4 + T - 1) / T, T, 0, stream>>>(
            (const float4*)h2, dinv, (const float4*)b2, (float4*)out, total4, OUTDIM / 4);
        k_scatter64<<<(E + wavesPerBlk - 1) / wavesPerBlk, T, 0, stream>>>(
            row, col, dinv, h2, out, E);
    }
}